// MIMOCell_14396730376485
// MI455X (gfx1250) — compile-verified
//
#include <hip/hip_runtime.h>
#include <hip/hip_bf16.h>

// MIMO decision-directed LMS equalizer scan (UP=DOWN=1 case).
// Strictly serial over 131072 steps -> single wave32 persistent kernel.
// Per-step 4x124 real matvec via 31 chained V_WMMA_F32_16X16X4_F32;
// LMS rank-1 update as packed v_pk_fma_f32 in the weight-owning lanes.

typedef float v2f __attribute__((ext_vector_type(2)));
typedef float v8f __attribute__((ext_vector_type(8)));

#define NUM_TAPS 31
#define MU_F     1e-3f
#define INV_SQRT2 0.70710678118654752f

__launch_bounds__(32)
__global__ void mimo_lms_scan_kernel(const float* __restrict__ xs,     // (N,2) complex64 -> 4 floats/step
                                     const float* __restrict__ w0,     // (2,2,31,1) complex64
                                     const float* __restrict__ fifo0,  // (31,2) complex64
                                     float* __restrict__ out,          // (N,2) complex64 -> 4 floats/step
                                     int nsteps)
{
    const int lane = threadIdx.x;        // 0..31
    const int h    = lane >> 4;          // half-wave selects q (A K-split low/high)
    const int m    = lane & 15;          // M row held by this lane in A layout
    const int mm   = m & 3;              // row within the 4-row group
    // Rows 0..3 hold the real expansion of W; rows 8..11 duplicate them so the
    // upper half-wave also receives y through the C/D layout (lanes 16-31 read M=8..15).
    const bool active = (m < 4) || (m >= 8 && m < 12);
    const int p = mm >> 1;               // output index (0 or 1)
    const int t = mm & 1;                // 0 = real-part row, 1 = imag-part row

    // A chunk k = flipped weights wk[p, q=h, k] = w0[p, h, 30-k].
    // Real expansion per chunk: re-row -> (wr, -wi) ; im-row -> (wi, wr).
    v2f a[NUM_TAPS];
    v2f b[NUM_TAPS];
#pragma unroll
    for (int k = 0; k < NUM_TAPS; ++k) {
        float wr = 0.0f, wi = 0.0f;
        if (active) {
            const int base = ((p * 2 + h) * NUM_TAPS + (NUM_TAPS - 1 - k)) * 2;
            wr = w0[base];
            wi = w0[base + 1];
        }
        v2f av;
        if (t) { av.x = wi; av.y = wr; }
        else   { av.x = wr; av.y = -wi; }
        a[k] = av;
        // FIFO init: fifo0[k, q=h] complex; replicated across the half-wave so
        // every B column equals the x-vector -> every D column equals y.
        v2f bv;
        bv.x = fifo0[k * 4 + h * 2 + 0];
        bv.y = fifo0[k * 4 + h * 2 + 1];
        b[k] = bv;
    }

    for (int i = 0; i < nsteps; ++i) {
        // stream xs ahead; locality 3 -> WGP-scope prefetch (near-cache fill),
        // uniform branch so EXEC stays all-ones at the WMMAs
        if ((i & 63) == 0)
            __builtin_prefetch(xs + (size_t)(i + 256) * 4, 0, 3);

        // FIFO shift (newest at tap 30); lowers to v_mov_b64 pairs
#pragma unroll
        for (int k = 0; k < NUM_TAPS - 1; ++k) b[k] = b[k + 1];
        {
            const float* xp = xs + (size_t)i * 4 + h * 2;  // coalesced broadcast per half
            v2f nv; nv.x = xp[0]; nv.y = xp[1];
            b[NUM_TAPS - 1] = nv;
        }

        // y = W_real(4x124) * x(124) as 31 chained f32 WMMAs (K=4 each, D->C accumulate)
        v8f c = {0.f, 0.f, 0.f, 0.f, 0.f, 0.f, 0.f, 0.f};
#pragma unroll
        for (int k = 0; k < NUM_TAPS; ++k) {
            c = __builtin_amdgcn_wmma_f32_16x16x4_f32(
                    /*neg_a=*/false, a[k],
                    /*neg_b=*/false, b[k],
                    /*c_mod=*/(short)0, c,
                    /*reuse_a=*/false, /*reuse_b=*/false);
        }
        // Row duplication means EVERY lane now holds y in c[0..3]:
        // lanes 0-15 read M=0..3, lanes 16-31 read M=8..11 (identical rows).
        const float y0r = c[0], y0i = c[1], y1r = c[2], y1i = c[3];

        // QPSK hard decision and error (computed redundantly in all lanes)
        const float e0r = (y0r >= 0.f ? INV_SQRT2 : -INV_SQRT2) - y0r;
        const float e0i = (y0i >= 0.f ? INV_SQRT2 : -INV_SQRT2) - y0i;
        const float e1r = (y1r >= 0.f ? INV_SQRT2 : -INV_SQRT2) - y1r;
        const float e1i = (y1i >= 0.f ? INV_SQRT2 : -INV_SQRT2) - y1i;

        // LMS: dwr = mu*(er*xr + ei*xi), dwi = mu*(ei*xr - er*xi)
        // re-row stores (wr, -wi): d(a) = ( mur, -mui)*xr + (mui, mur)*xi
        // im-row stores (wi,  wr): d(a) = ( mui,  mur)*xr + (-mur, mui)*xi
        const float er  = p ? e1r : e0r;
        const float ei  = p ? e1i : e0i;
        const float mur = active ? MU_F * er : 0.0f;   // inactive lanes: zero update,
        const float mui = active ? MU_F * ei : 0.0f;   // keeps control flow uniform
        v2f gx, gy;
        if (t) { gx.x = mui;  gx.y = mur; gy.x = -mur; gy.y = mui; }
        else   { gx.x = mur;  gx.y = -mui; gy.x = mui; gy.y = mur; }
#pragma unroll
        for (int k = 0; k < NUM_TAPS; ++k) {
            v2f xr2; xr2.x = b[k].x; xr2.y = b[k].x;
            v2f xi2; xi2.x = b[k].y; xi2.y = b[k].y;
            // packed 2xf32 FMA pair -> v_pk_fma_f32 (VOP3P)
            a[k] = __builtin_elementwise_fma(gx, xr2,
                   __builtin_elementwise_fma(gy, xi2, a[k]));
        }

        // emit y for this step: (y0.re, y0.im, y1.re, y1.im)
        if (lane == 0) {
            float4 o;
            o.x = y0r; o.y = y0i; o.z = y1r; o.w = y1i;
            *(float4*)(out + (size_t)i * 4) = o;
        }
    }
}

extern "C" void kernel_launch(void* const* d_in, const int* in_sizes, int n_in,
                              void* d_out, int out_size, void* d_ws, size_t ws_size,
                              hipStream_t stream) {
    const float* xs    = (const float*)d_in[0];   // complex64 viewed as float pairs
    const float* w0    = (const float*)d_in[1];
    const float* fifo0 = (const float*)d_in[2];
    float*       out   = (float*)d_out;

    // output is (N_STEPS, 2) complex64 -> 4 floats per step
    const int nsteps = out_size / 4;

    hipLaunchKernelGGL(mimo_lms_scan_kernel, dim3(1), dim3(32), 0, stream,
                       xs, w0, fifo0, out, nsteps);
}